// AXRFeatureLoss_20100446945623
// MI455X (gfx1250) — compile-verified
//
#include <hip/hip_runtime.h>
#include <math.h>

#define HWs 9216
#define Hs 96
#define Wsz 96
#define Bs 8
#define Cs 256
#define Kcls 6
#define NEG_INF (-__builtin_inff())

typedef __attribute__((ext_vector_type(2))) float v2f;
typedef __attribute__((ext_vector_type(8))) float v8f;

// ---------------- workspace layout (float offsets) ----------------
static constexpr size_t OFF_STATS = 0;                                   // 2 tensors * (scale[256], shift[256])
static constexpr size_t OFF_WCLS  = OFF_STATS + 1024;                    // 2*6*256
static constexpr size_t OFF_BCLS  = OFF_WCLS + (size_t)2*Kcls*Cs;        // 2*16 (padded)
static constexpr size_t OFF_WQKV  = OFF_BCLS + 32;                       // 2*320*256 (rows 0-31 q', 32-63 k', 64-319 v')
static constexpr size_t OFF_BQKV  = OFF_WQKV + (size_t)2*320*Cs;         // 2*320
static constexpr size_t OFF_Y     = OFF_BQKV + 2*320;                    // 2*8*320*9216 (q|k|v activations)
static constexpr size_t OFF_ATT   = OFF_Y   + (size_t)2*Bs*320*HWs;      // 2*8*96*96*192
static constexpr size_t OFF_D     = OFF_ATT + (size_t)2*Bs*Hs*Wsz*192;   // 8*256*9216 (out diff accumulator)
static constexpr size_t OFF_M     = OFF_D   + (size_t)Bs*Cs*HWs;         // 2*8*6*9216 (cls logits/softmax)
static constexpr size_t OFF_CA    = OFF_M   + (size_t)2*Bs*Kcls*HWs;     // 2*8*6*256
static constexpr size_t OFF_ACC   = OFF_CA  + (size_t)2*Bs*Kcls*Cs;      // [0]=cc_sumsq [1]=ca_sumsq

__device__ __forceinline__ v8f wmma4(v2f a, v2f b, v8f c) {
    return __builtin_amdgcn_wmma_f32_16x16x4_f32(false, a, false, b, (short)0, c, false, false);
}

// async global -> LDS copy of 16 bytes (gfx1250). For generic pointers into the
// LDS aperture, the LDS address is the low 32 bits of the pointer (ISA 10.2), so truncating the flat
// __shared__ pointer yields the per-wave LDS byte address the VDST operand wants.
__device__ __forceinline__ void async_g2l_b128(void* lds_dst, const void* gsrc) {
    asm volatile("global_load_async_to_lds_b128 %0, %1, off"
                 :: "v"((unsigned)(unsigned long long)lds_dst),
                    "v"((unsigned long long)gsrc)
                 : "memory");
}
__device__ __forceinline__ void wait_asynccnt0() {
#if __has_builtin(__builtin_amdgcn_s_wait_asynccnt)
    __builtin_amdgcn_s_wait_asynccnt(0);
#else
    asm volatile("s_wait_asynccnt 0x0" ::: "memory");
#endif
}

// ---------------- zero scratch region ----------------
__global__ void k_zero(float* p, long long n) {
    long long i = (long long)blockIdx.x * blockDim.x + threadIdx.x;
    long long st = (long long)gridDim.x * blockDim.x;
    for (; i < n; i += st) p[i] = 0.f;
}

// ---------------- per-channel mean / ddof=1 std -> scale, shift ----------------
__global__ void k_stats(const float* S, const float* T, float* ws) {
    int c = blockIdx.x;
    const float* X = blockIdx.y ? T : S;
    float sum = 0.f, ss = 0.f;
    for (int n = 0; n < Bs; ++n) {
        const float* p = X + ((size_t)n * Cs + c) * HWs;
        for (int i = threadIdx.x; i < HWs; i += 256) { float v = p[i]; sum += v; ss += v * v; }
    }
    __shared__ float s1[256], s2[256];
    s1[threadIdx.x] = sum; s2[threadIdx.x] = ss; __syncthreads();
    for (int o = 128; o > 0; o >>= 1) {
        if (threadIdx.x < o) { s1[threadIdx.x] += s1[threadIdx.x + o]; s2[threadIdx.x] += s2[threadIdx.x + o]; }
        __syncthreads();
    }
    if (threadIdx.x == 0) {
        float nTot  = (float)(Bs * HWs);
        float mean  = s1[0] / nTot;
        float var   = (s2[0] - nTot * mean * mean) / (nTot - 1.0f);
        float stdd  = sqrtf(fmaxf(var, 0.f));
        float scale = 1.f / (stdd + 1e-6f);
        float* base = ws + OFF_STATS + (size_t)blockIdx.y * 512;
        base[c] = scale; base[256 + c] = -mean * scale;
    }
}

// ---------------- fold normalization into conv weights ----------------
__global__ void k_fold(const float* wcls, const float* wq, const float* bq,
                       const float* wk, const float* bk, const float* wv, const float* bv,
                       float* ws) {
    int r = blockIdx.x;           // 0..325
    int t = blockIdx.y;           // 0=S, 1=T
    const float* scale = ws + OFF_STATS + (size_t)t * 512;
    const float* shift = scale + 256;
    const float* wsrc; float bsrc = 0.f; float* wdst; float* bdst;
    if (r < Kcls) {
        wsrc = wcls + (size_t)r * Cs;
        wdst = ws + OFF_WCLS + ((size_t)t * Kcls + r) * Cs;
        bdst = ws + OFF_BCLS + (size_t)t * 16 + r;
    } else {
        int rr = r - Kcls;        // 0..319
        if (rr < 32)       { wsrc = wq + (size_t)rr * Cs;        bsrc = bq[rr]; }
        else if (rr < 64)  { wsrc = wk + (size_t)(rr - 32) * Cs; bsrc = bk[rr - 32]; }
        else               { wsrc = wv + (size_t)(rr - 64) * Cs; bsrc = bv[rr - 64]; }
        wdst = ws + OFF_WQKV + ((size_t)t * 320 + rr) * Cs;
        bdst = ws + OFF_BQKV + (size_t)t * 320 + rr;
    }
    int c = threadIdx.x;
    float w = wsrc[c];
    wdst[c] = w * scale[c];
    __shared__ float sred[256];
    sred[c] = w * shift[c]; __syncthreads();
    for (int o = 128; o > 0; o >>= 1) { if (c < o) sred[c] += sred[c + o]; __syncthreads(); }
    if (c == 0) *bdst = bsrc + sred[0];
}

// ---------------- WMMA GEMM: Y[t][n][o][p] = W'(320x256) @ X(256x9216) + b' ----------------
__global__ void k_gemm_qkv(const float* S, const float* T, float* ws) {
    int wave = threadIdx.x >> 5, lane = threadIdx.x & 31;
    int m16 = lane & 15, kh = lane >> 4;
    int ptile = (blockIdx.x * 8 + wave) * 16;
    int otile = blockIdx.y * 16;
    int z = blockIdx.z; int t = z >> 3; int n = z & 7;
    const float* Wf = ws + OFF_WQKV + (size_t)t * 320 * Cs;
    const float* bf = ws + OFF_BQKV + (size_t)t * 320;
    const float* X  = (t ? T : S) + (size_t)n * Cs * HWs;
    float* Y = ws + OFF_Y + ((size_t)t * Bs + n) * 320 * HWs;
    const float* Arow = Wf + (size_t)(otile + m16) * Cs + 2 * kh;
    const float* Bcol = X + (size_t)(2 * kh) * HWs + ptile + m16;
    v8f acc = {};
    #pragma unroll 8
    for (int k0 = 0; k0 < Cs; k0 += 4) {
        v2f a; a.x = Arow[k0];                 a.y = Arow[k0 + 1];
        v2f b; b.x = Bcol[(size_t)k0 * HWs];   b.y = Bcol[(size_t)(k0 + 1) * HWs];
        acc = wmma4(a, b, acc);
    }
    #pragma unroll
    for (int r = 0; r < 8; ++r) {
        int o = otile + r + 8 * kh;
        Y[(size_t)o * HWs + ptile + m16] = acc[r] + bf[o];
    }
}

// ---------------- eH gram (fixed y): E[i,j] = sum_c q[c,i,y]*k[c,j,y], diag -> -inf ----------------
__global__ void k_eH(float* ws) {
    int lane = threadIdx.x; int m16 = lane & 15, kh = lane >> 4;
    int ti = blockIdx.x * 16, tj = blockIdx.y * 16;
    int z = blockIdx.z; int y = z % Wsz; int n = (z / Wsz) & 7; int t = z / (Wsz * Bs);
    const float* Yb = ws + OFF_Y + ((size_t)t * Bs + n) * 320 * HWs;
    const float* Q  = Yb;
    const float* Km = Yb + (size_t)32 * HWs;
    const float* Ap = Q  + (size_t)(2 * kh) * HWs + (size_t)(ti + m16) * Wsz + y;
    const float* Bp = Km + (size_t)(2 * kh) * HWs + (size_t)(tj + m16) * Wsz + y;
    v8f acc = {};
    #pragma unroll
    for (int k0 = 0; k0 < 32; k0 += 4) {
        v2f a; a.x = Ap[(size_t)k0 * HWs]; a.y = Ap[(size_t)(k0 + 1) * HWs];
        v2f b; b.x = Bp[(size_t)k0 * HWs]; b.y = Bp[(size_t)(k0 + 1) * HWs];
        acc = wmma4(a, b, acc);
    }
    float* att = ws + OFF_ATT + ((size_t)t * Bs + n) * Hs * Wsz * 192;
    #pragma unroll
    for (int r = 0; r < 8; ++r) {
        int x = ti + r + 8 * kh, j = tj + m16;
        att[(size_t)x * (Wsz * 192) + (size_t)y * 192 + j] = (x == j) ? NEG_INF : acc[r];
    }
}

// ---------------- eW gram (fixed x): E[i,j] = sum_c q[c,x,i]*k[c,x,j] ----------------
__global__ void k_eW(float* ws) {
    int lane = threadIdx.x; int m16 = lane & 15, kh = lane >> 4;
    int ti = blockIdx.x * 16, tj = blockIdx.y * 16;
    int z = blockIdx.z; int x = z % Hs; int n = (z / Hs) & 7; int t = z / (Hs * Bs);
    const float* Yb = ws + OFF_Y + ((size_t)t * Bs + n) * 320 * HWs;
    const float* Q  = Yb;
    const float* Km = Yb + (size_t)32 * HWs;
    const float* Ap = Q  + (size_t)(2 * kh) * HWs + (size_t)x * Wsz + ti + m16;
    const float* Bp = Km + (size_t)(2 * kh) * HWs + (size_t)x * Wsz + tj + m16;
    v8f acc = {};
    #pragma unroll
    for (int k0 = 0; k0 < 32; k0 += 4) {
        v2f a; a.x = Ap[(size_t)k0 * HWs]; a.y = Ap[(size_t)(k0 + 1) * HWs];
        v2f b; b.x = Bp[(size_t)k0 * HWs]; b.y = Bp[(size_t)(k0 + 1) * HWs];
        acc = wmma4(a, b, acc);
    }
    float* att = ws + OFF_ATT + ((size_t)t * Bs + n) * Hs * Wsz * 192;
    #pragma unroll
    for (int r = 0; r < 8; ++r) {
        int i = ti + r + 8 * kh, j = tj + m16;
        att[(size_t)x * (Wsz * 192) + (size_t)i * 192 + 96 + j] = acc[r];
    }
}

// ---------------- softmax over rows of 192 (one wave per row, 6 elems/lane) ----------------
__global__ void k_softmax_att(float* ws) {
    int wave = threadIdx.x >> 5, lane = threadIdx.x & 31;
    long long row = (long long)blockIdx.x * 8 + wave;          // 2*8*96*96 rows total
    float* p = ws + OFF_ATT + row * 192;
    float v[6]; float mx = NEG_INF;
    #pragma unroll
    for (int e = 0; e < 6; ++e) { v[e] = p[lane + 32 * e]; mx = fmaxf(mx, v[e]); }
    #pragma unroll
    for (int o = 16; o > 0; o >>= 1) mx = fmaxf(mx, __shfl_xor(mx, o, 32));
    float sum = 0.f;
    #pragma unroll
    for (int e = 0; e < 6; ++e) { v[e] = __expf(v[e] - mx); sum += v[e]; }
    #pragma unroll
    for (int o = 16; o > 0; o >>= 1) sum += __shfl_xor(sum, o, 32);
    float inv = 1.f / sum;
    #pragma unroll
    for (int e = 0; e < 6; ++e) p[lane + 32 * e] = v[e] * inv;
}

// ---------------- outH/outW WMMA (K=96), async-staged B panel in LDS ----------------
// Block = 256 threads (8 waves). One block handles all 16 channel tiles of a
// (t, n, s, itile) slice; the shared 16x96 attention B-panel is staged into LDS
// once via GLOBAL_LOAD_ASYNC_TO_LDS_B128 and reused by every wave/tile.
__global__ void k_out_attn(float* ws, int mode) {
    __shared__ float tileB[16 * 100];     // [nn][kk], rows padded to 100 floats (400B, 16B-aligned)
    int tid = threadIdx.x; int wave = tid >> 5, lane = tid & 31;
    int m16 = lane & 15, kh = lane >> 4;
    int itile = blockIdx.x * 16;          // spatial output tile
    int s = blockIdx.y;                   // fixed y (mode 0) or fixed x (mode 1)
    int z = blockIdx.z; int t = z >> 3; int n = z & 7;
    const float* Yb  = ws + OFF_Y + ((size_t)t * Bs + n) * 320 * HWs;
    const float* V   = Yb + (size_t)64 * HWs;
    const float* att = ws + OFF_ATT + ((size_t)t * Bs + n) * Hs * Wsz * 192;
    float sign = t ? 1.f : -1.f;

    // stage B[kk][nn]: mode0 = att[(itile+nn)*W*192 + s*192 + kk]
    //                  mode1 = att[s*W*192 + (itile+nn)*192 + 96 + kk]
    // (contiguous in kk for fixed nn: 96 floats = 24 x b128 per row, 384 chunks)
    for (int q = tid; q < 384; q += 256) {
        int nn = q / 24, ch = q % 24;
        const float* src = (mode == 0)
            ? att + (size_t)(itile + nn) * (Wsz * 192) + (size_t)s * 192 + ch * 4
            : att + (size_t)s * (Wsz * 192) + (size_t)(itile + nn) * 192 + 96 + ch * 4;
        async_g2l_b128(&tileB[nn * 100 + ch * 4], src);
    }
    wait_asynccnt0();
    __syncthreads();

    float* D = ws + OFF_D;
    #pragma unroll
    for (int rep = 0; rep < 2; ++rep) {
        int ctile = (wave * 2 + rep) * 16;
        const float* Ap = (mode == 0)
            ? V + (size_t)(ctile + m16) * HWs + (size_t)(2 * kh) * Wsz + s
            : V + (size_t)(ctile + m16) * HWs + (size_t)s * Wsz + 2 * kh;
        size_t aStep = (mode == 0) ? (size_t)Wsz : (size_t)1;
        const float* Bl = &tileB[m16 * 100 + 2 * kh];
        v8f acc = {};
        #pragma unroll 6
        for (int k0 = 0; k0 < 96; k0 += 4) {
            v2f a; a.x = Ap[(size_t)k0 * aStep]; a.y = Ap[(size_t)(k0 + 1) * aStep];
            v2f b; b.x = Bl[k0];                 b.y = Bl[k0 + 1];
            acc = wmma4(a, b, acc);
        }
        size_t outBase = (mode == 0)
            ? (size_t)n * Cs * HWs + (size_t)(itile + m16) * Wsz + s
            : (size_t)n * Cs * HWs + (size_t)s * Wsz + itile + m16;
        #pragma unroll
        for (int r = 0; r < 8; ++r) {
            int c = ctile + r + 8 * kh;
            atomicAdd(&D[outBase + (size_t)c * HWs], sign * acc[r]);
        }
    }
}

// ---------------- causal-attention: class logits ----------------
__global__ void k_cls(const float* S, const float* T, float* ws) {
    int p = blockIdx.x * 256 + threadIdx.x;
    int n = blockIdx.y, t = blockIdx.z;
    const float* X  = (t ? T : S) + (size_t)n * Cs * HWs + p;
    const float* Wc = ws + OFF_WCLS + (size_t)t * Kcls * Cs;
    const float* bc = ws + OFF_BCLS + (size_t)t * 16;
    float acc[Kcls];
    #pragma unroll
    for (int k = 0; k < Kcls; ++k) acc[k] = bc[k];
    for (int c = 0; c < Cs; ++c) {
        float xv = X[(size_t)c * HWs];
        #pragma unroll
        for (int k = 0; k < Kcls; ++k) acc[k] = fmaf(Wc[(size_t)k * Cs + c], xv, acc[k]);
    }
    float* M = ws + OFF_M + ((size_t)(t * Bs + n) * Kcls) * HWs + p;
    #pragma unroll
    for (int k = 0; k < Kcls; ++k) M[(size_t)k * HWs] = acc[k];
}

// ---------------- softmax over 9216-wide rows (96 rows) ----------------
__global__ void k_softmaxM(float* ws) {
    float* row = ws + OFF_M + (size_t)blockIdx.x * HWs;
    __shared__ float sred[256];
    float mx = NEG_INF;
    for (int i = threadIdx.x; i < HWs; i += 256) mx = fmaxf(mx, row[i]);
    sred[threadIdx.x] = mx; __syncthreads();
    for (int o = 128; o > 0; o >>= 1) {
        if (threadIdx.x < o) sred[threadIdx.x] = fmaxf(sred[threadIdx.x], sred[threadIdx.x + o]);
        __syncthreads();
    }
    mx = sred[0]; __syncthreads();
    float sum = 0.f;
    for (int i = threadIdx.x; i < HWs; i += 256) { float e = __expf(row[i] - mx); row[i] = e; sum += e; }
    sred[threadIdx.x] = sum; __syncthreads();
    for (int o = 128; o > 0; o >>= 1) { if (threadIdx.x < o) sred[threadIdx.x] += sred[threadIdx.x + o]; __syncthreads(); }
    float inv = 1.f / sred[0];
    for (int i = threadIdx.x; i < HWs; i += 256) row[i] *= inv;
}

// ---------------- ca[n,k,c] = scale_c * <sm, x_c> + shift_c  (softmax sums to 1) ----------------
__global__ void k_ca(const float* S, const float* T, float* ws) {
    int c = blockIdx.x; int nk = blockIdx.y; int t = blockIdx.z;
    int n = nk / Kcls, k = nk % Kcls;
    const float* X  = (t ? T : S) + ((size_t)n * Cs + c) * HWs;
    const float* sm = ws + OFF_M + ((size_t)(t * Bs + n) * Kcls + k) * HWs;
    float dot = 0.f;
    for (int i = threadIdx.x; i < HWs; i += 256) dot += sm[i] * X[i];
    __shared__ float sred[256];
    sred[threadIdx.x] = dot; __syncthreads();
    for (int o = 128; o > 0; o >>= 1) { if (threadIdx.x < o) sred[threadIdx.x] += sred[threadIdx.x + o]; __syncthreads(); }
    if (threadIdx.x == 0) {
        const float* st = ws + OFF_STATS + (size_t)t * 512;
        ws[OFF_CA + ((size_t)(t * Bs + n) * Kcls + k) * Cs + c] = st[c] * sred[0] + st[256 + c];
    }
}

// ---------------- ca loss: sum (ca_t - ca_s)^2 ----------------
__global__ void k_caloss(float* ws) {
    const float* caS = ws + OFF_CA;
    const float* caT = ws + OFF_CA + (size_t)Bs * Kcls * Cs;
    float acc = 0.f;
    for (int i = threadIdx.x; i < Bs * Kcls * Cs; i += 256) { float d = caT[i] - caS[i]; acc += d * d; }
    __shared__ float sred[256];
    sred[threadIdx.x] = acc; __syncthreads();
    for (int o = 128; o > 0; o >>= 1) { if (threadIdx.x < o) sred[threadIdx.x] += sred[threadIdx.x + o]; __syncthreads(); }
    if (threadIdx.x == 0) ws[OFF_ACC + 1] = sred[0];
}

// ---------------- cc loss: sum (gamma*D + t_norm - s_norm)^2 ----------------
__global__ void k_ccreduce(const float* S, const float* T, const float* gamma, float* ws) {
    const float* D   = ws + OFF_D;
    const float* stS = ws + OFF_STATS;
    const float* stT = ws + OFF_STATS + 512;
    float g = gamma[0];
    long long total = (long long)Bs * Cs * HWs;
    float acc = 0.f;
    for (long long i = (long long)blockIdx.x * blockDim.x + threadIdx.x; i < total;
         i += (long long)gridDim.x * blockDim.x) {
        int c = (int)((i / HWs) % Cs);
        float tn = T[i] * stT[c] + stT[256 + c];
        float sn = S[i] * stS[c] + stS[256 + c];
        float d  = g * D[i] + tn - sn;
        acc += d * d;
    }
    __shared__ float sred[256];
    sred[threadIdx.x] = acc; __syncthreads();
    for (int o = 128; o > 0; o >>= 1) { if (threadIdx.x < o) sred[threadIdx.x] += sred[threadIdx.x + o]; __syncthreads(); }
    if (threadIdx.x == 0) atomicAdd(&ws[OFF_ACC], sred[0]);
}

__global__ void k_final(const float* ws, float* out) {
    out[0] = ws[OFF_ACC + 1] * (0.0005f / (float)Bs) + ws[OFF_ACC] * (1e-5f / (float)Bs);
}

extern "C" void kernel_launch(void* const* d_in, const int* in_sizes, int n_in,
                              void* d_out, int out_size, void* d_ws, size_t ws_size,
                              hipStream_t stream) {
    (void)in_sizes; (void)n_in; (void)out_size; (void)ws_size;
    const float* S     = (const float*)d_in[0];
    const float* T     = (const float*)d_in[1];
    const float* wcls  = (const float*)d_in[2];
    const float* wq    = (const float*)d_in[3];
    const float* bq    = (const float*)d_in[4];
    const float* wk    = (const float*)d_in[5];
    const float* bk    = (const float*)d_in[6];
    const float* wv    = (const float*)d_in[7];
    const float* bv    = (const float*)d_in[8];
    const float* gamma = (const float*)d_in[9];
    float* ws = (float*)d_ws;
    float* out = (float*)d_out;

    // zero D accumulator + loss accumulators (idempotent per call)
    k_zero<<<2048, 256, 0, stream>>>(ws + OFF_D, (long long)(OFF_ACC + 8 - OFF_D));
    k_stats<<<dim3(256, 2), 256, 0, stream>>>(S, T, ws);
    k_fold<<<dim3(326, 2), 256, 0, stream>>>(wcls, wq, bq, wk, bk, wv, bv, ws);
    k_gemm_qkv<<<dim3(72, 20, 16), 256, 0, stream>>>(S, T, ws);
    k_eH<<<dim3(6, 6, 2 * Bs * Wsz), 32, 0, stream>>>(ws);
    k_eW<<<dim3(6, 6, 2 * Bs * Hs), 32, 0, stream>>>(ws);
    k_softmax_att<<<(2 * Bs * Hs * Wsz) / 8, 256, 0, stream>>>(ws);
    k_out_attn<<<dim3(6, 96, 16), 256, 0, stream>>>(ws, 0);
    k_out_attn<<<dim3(6, 96, 16), 256, 0, stream>>>(ws, 1);
    k_cls<<<dim3(HWs / 256, Bs, 2), 256, 0, stream>>>(S, T, ws);
    k_softmaxM<<<2 * Bs * Kcls, 256, 0, stream>>>(ws);
    k_ca<<<dim3(Cs, Bs * Kcls, 2), 256, 0, stream>>>(S, T, ws);
    k_caloss<<<1, 256, 0, stream>>>(ws);
    k_ccreduce<<<1024, 256, 0, stream>>>(S, T, gamma, ws);
    k_final<<<1, 1, 0, stream>>>(ws, out);
}